// CompGCNLayer_944892805204
// MI455X (gfx1250) — compile-verified
//
#include <hip/hip_runtime.h>

typedef float v2f __attribute__((ext_vector_type(2)));
typedef float v8f __attribute__((ext_vector_type(8)));

#define D_FEAT 128

// ---------------------------------------------------------------------------
// Phase 1: edge scatter.  One wave32 per edge; each lane owns 4 consecutive
// floats (32 lanes * 4 = 128).  msg_o = x[src]-e -> atomic add at dst,
// msg_i = x[dst]-e -> atomic add at src.  Lane 0 counts degrees.
// ---------------------------------------------------------------------------
__global__ __launch_bounds__(256) void edge_scatter_kernel(
    const float* __restrict__ node, const float* __restrict__ edge,
    const int* __restrict__ src, const int* __restrict__ dst,
    float* __restrict__ ho_sum, float* __restrict__ hi_sum,
    float* __restrict__ deg_o, float* __restrict__ deg_i, int n_edges)
{
    const int lane = threadIdx.x & 31;
    const int e = blockIdx.x * (blockDim.x >> 5) + (threadIdx.x >> 5);
    if (e >= n_edges) return;

    const int s = src[e];
    const int d = dst[e];

    const float4 xs = *(const float4*)(node + (size_t)s * D_FEAT + lane * 4);
    const float4 xd = *(const float4*)(node + (size_t)d * D_FEAT + lane * 4);
    const float4 ee = *(const float4*)(edge + (size_t)e * D_FEAT + lane * 4);

    float* po = ho_sum + (size_t)d * D_FEAT + lane * 4;
    unsafeAtomicAdd(po + 0, xs.x - ee.x);
    unsafeAtomicAdd(po + 1, xs.y - ee.y);
    unsafeAtomicAdd(po + 2, xs.z - ee.z);
    unsafeAtomicAdd(po + 3, xs.w - ee.w);

    float* pi = hi_sum + (size_t)s * D_FEAT + lane * 4;
    unsafeAtomicAdd(pi + 0, xd.x - ee.x);
    unsafeAtomicAdd(pi + 1, xd.y - ee.y);
    unsafeAtomicAdd(pi + 2, xd.z - ee.z);
    unsafeAtomicAdd(pi + 3, xd.w - ee.w);

    if (lane == 0) {
        unsafeAtomicAdd(deg_o + d, 1.0f);
        unsafeAtomicAdd(deg_i + s, 1.0f);
    }
}

// ---------------------------------------------------------------------------
// Phase 2: fused triple GEMM (fp32 WMMA 16x16x4) + bias + /3 + BN partial
// sums.  One wave per 16x16 output tile; block of 8 waves covers one m-tile
// (16 nodes) across all 8 n-tiles (128 feats) so A rows are L1-shared.
//
// ISA fragment layouts (wave32):
//   A (16x4 f32, 2 VGPRs): lane L -> row M = L&15, K = vgpr + 2*(L>>4)
//   B (4x16 f32, 2 VGPRs): lane L -> col N = L&15, K = vgpr + 2*(L>>4)
//   C/D (16x16 f32, 8 VGPRs): lane L -> col N = L&15, row M = vgpr + 8*(L>>4)
// h[n,o] = sum_k in[n,k] * W[o,k]  (weights are [out,in]) so B[k][o]=W[o][k].
// ---------------------------------------------------------------------------
__global__ __launch_bounds__(256) void gemm_bn_partial_kernel(
    const float* __restrict__ ho_sum, const float* __restrict__ hi_sum,
    const float* __restrict__ x,
    const float* __restrict__ deg_o, const float* __restrict__ deg_i,
    const float* __restrict__ W_O, const float* __restrict__ W_I,
    const float* __restrict__ W_S,
    const float* __restrict__ b_O, const float* __restrict__ b_I,
    const float* __restrict__ b_S,
    float* __restrict__ h_out,
    float* __restrict__ colsum, float* __restrict__ colsumsq)
{
    const int lane = threadIdx.x & 31;
    const int wv   = threadIdx.x >> 5;
    const int tile = blockIdx.x * 8 + wv;     // tm = blockIdx.x, tn = wv
    const int tm   = tile >> 3;
    const int tn   = tile & 7;
    const int m    = lane & 15;
    const int half = lane >> 4;

    const int node = tm * 16 + m;             // A row owned by this lane
    const int ncol = tn * 16 + m;             // B/D column owned by this lane

    const float inv_o = 1.0f / fmaxf(deg_o[node], 1.0f);
    const float inv_i = 1.0f / fmaxf(deg_i[node], 1.0f);

    const float* ro = ho_sum + (size_t)node * D_FEAT;
    const float* ri = hi_sum + (size_t)node * D_FEAT;
    const float* rx = x      + (size_t)node * D_FEAT;
    const float* wo = W_O + (size_t)ncol * D_FEAT;
    const float* wi = W_I + (size_t)ncol * D_FEAT;
    const float* wsrow = W_S + (size_t)ncol * D_FEAT;

    v8f acc = {};
    #pragma unroll 4
    for (int k0 = 0; k0 < D_FEAT; k0 += 4) {
        const int ka = k0 + 2 * half;         // this lane's K base
        v2f a, b;

        a[0] = ro[ka] * inv_o;  a[1] = ro[ka + 1] * inv_o;
        b[0] = wo[ka];          b[1] = wo[ka + 1];
        acc = __builtin_amdgcn_wmma_f32_16x16x4_f32(false, a, false, b,
                                                    (short)0, acc, false, false);

        a[0] = ri[ka] * inv_i;  a[1] = ri[ka + 1] * inv_i;
        b[0] = wi[ka];          b[1] = wi[ka + 1];
        acc = __builtin_amdgcn_wmma_f32_16x16x4_f32(false, a, false, b,
                                                    (short)0, acc, false, false);

        a[0] = rx[ka];          a[1] = rx[ka + 1];
        b[0] = wsrow[ka];       b[1] = wsrow[ka + 1];
        acc = __builtin_amdgcn_wmma_f32_16x16x4_f32(false, a, false, b,
                                                    (short)0, acc, false, false);
    }

    const float bsum = b_O[ncol] + b_I[ncol] + b_S[ncol];
    const int rowbase = tm * 16 + 8 * half;

    float s = 0.0f, ss = 0.0f;
    #pragma unroll
    for (int r = 0; r < 8; ++r) {
        const float v = (acc[r] + bsum) * (1.0f / 3.0f);
        h_out[(size_t)(rowbase + r) * D_FEAT + ncol] = v;
        s  += v;
        ss += v * v;
    }

    // lane L and lane L^16 own the same output column -> combine then one
    // atomic per column per wave.
    s  += __shfl_xor(s, 16, 32);
    ss += __shfl_xor(ss, 16, 32);
    if (half == 0) {
        unsafeAtomicAdd(colsum   + ncol, s);
        unsafeAtomicAdd(colsumsq + ncol, ss);
    }
}

// ---------------------------------------------------------------------------
// Phase 3a: per-feature BN scale/shift (training-mode batch stats).
// ---------------------------------------------------------------------------
__global__ void bn_stats_kernel(const float* __restrict__ colsum,
                                const float* __restrict__ colsumsq,
                                const float* __restrict__ gamma,
                                const float* __restrict__ beta,
                                float* __restrict__ scale,
                                float* __restrict__ shift, float inv_n)
{
    const int j = threadIdx.x;
    const float mu  = colsum[j] * inv_n;
    const float var = colsumsq[j] * inv_n - mu * mu;
    const float sc  = gamma[j] * rsqrtf(var + 1e-5f);
    scale[j] = sc;
    shift[j] = beta[j] - mu * sc;
}

// ---------------------------------------------------------------------------
// Phase 3b: in-place affine over d_out (float4, columns aligned mod 128).
// ---------------------------------------------------------------------------
__global__ __launch_bounds__(256) void bn_apply_kernel(
    float* __restrict__ out, const float* __restrict__ scale,
    const float* __restrict__ shift, int total4)
{
    const int t = blockIdx.x * blockDim.x + threadIdx.x;
    if (t >= total4) return;
    const int j = (t * 4) & (D_FEAT - 1);
    float4 v = ((float4*)out)[t];
    const float4 sc = *(const float4*)(scale + j);
    const float4 sh = *(const float4*)(shift + j);
    v.x = v.x * sc.x + sh.x;
    v.y = v.y * sc.y + sh.y;
    v.z = v.z * sc.z + sh.z;
    v.w = v.w * sc.w + sh.w;
    ((float4*)out)[t] = v;
}

// ---------------------------------------------------------------------------
extern "C" void kernel_launch(void* const* d_in, const int* in_sizes, int n_in,
                              void* d_out, int out_size, void* d_ws, size_t ws_size,
                              hipStream_t stream)
{
    const float* node  = (const float*)d_in[0];
    const float* edge  = (const float*)d_in[1];
    const float* W_O   = (const float*)d_in[2];
    const float* b_O   = (const float*)d_in[3];
    const float* W_I   = (const float*)d_in[4];
    const float* b_I   = (const float*)d_in[5];
    const float* W_S   = (const float*)d_in[6];
    const float* b_S   = (const float*)d_in[7];
    const float* gamma = (const float*)d_in[8];
    const float* beta  = (const float*)d_in[9];
    const int*   src   = (const int*)d_in[10];
    const int*   dst   = (const int*)d_in[11];

    const int N = in_sizes[0] / D_FEAT;   // 100000 (divisible by 16)
    const int E = in_sizes[1] / D_FEAT;   // 600000

    float* ws = (float*)d_ws;
    const size_t ND = (size_t)N * D_FEAT;
    float* ho_sum   = ws;
    float* hi_sum   = ws + ND;
    float* deg_o    = ws + 2 * ND;
    float* deg_i    = deg_o + N;
    float* colsum   = deg_i + N;
    float* colsumsq = colsum + D_FEAT;
    float* scale    = colsumsq + D_FEAT;
    float* shift    = scale + D_FEAT;

    // zero all accumulators (graph-capture-safe memset node)
    const size_t zero_bytes = (2 * ND + 2 * (size_t)N + 4 * D_FEAT) * sizeof(float);
    hipMemsetAsync(d_ws, 0, zero_bytes, stream);

    // Phase 1: scatter (8 edges per 256-thread block)
    const int eblocks = (E + 7) / 8;
    edge_scatter_kernel<<<eblocks, 256, 0, stream>>>(
        node, edge, src, dst, ho_sum, hi_sum, deg_o, deg_i, E);

    // Phase 2: WMMA GEMM + BN partials (one block per m-tile, 8 waves = 8 n-tiles)
    const int mtiles = N / 16;
    gemm_bn_partial_kernel<<<mtiles, 256, 0, stream>>>(
        ho_sum, hi_sum, node, deg_o, deg_i, W_O, W_I, W_S, b_O, b_I, b_S,
        (float*)d_out, colsum, colsumsq);

    // Phase 3: BN finalize + apply
    bn_stats_kernel<<<1, D_FEAT, 0, stream>>>(colsum, colsumsq, gamma, beta,
                                              scale, shift, 1.0f / (float)N);
    const int total4 = (int)(ND / 4);
    bn_apply_kernel<<<(total4 + 255) / 256, 256, 0, stream>>>(
        (float*)d_out, scale, shift, total4);
}